// HierarchicalGNNBlock_64166811402583
// MI455X (gfx1250) — compile-verified
//
#include <hip/hip_runtime.h>
#include <math.h>

// ---------------------------------------------------------------------------
// Problem constants (from reference hparams)
// ---------------------------------------------------------------------------
#define NC      1024      // clusters
#define EMB     16
#define LATENT  128
#define HIDDEN  512
#define K_SUPER 10
#define K_BIP   8
#define EPSF    1e-12f

typedef __attribute__((ext_vector_type(2))) float v2f;
typedef __attribute__((ext_vector_type(8))) float v8f;

// ---------------------------------------------------------------------------
// Workspace layout (float offsets). Accumulators first so one zero-fill
// covers them every call (harness does not re-zero between replays).
// ---------------------------------------------------------------------------
#define WS_CSUMS   0                         // [NC*EMB]   cluster emb sums
#define WS_CNT     (WS_CSUMS + NC*EMB)       // [NC]       cluster counts
#define WS_BNS     (WS_CNT + NC)             // [16]       bip BN sum / sumsq
#define WS_BDEN    (WS_BNS + 16)             // [NC]       bip denom
#define WS_SNIN    (WS_BDEN + NC)            // [NC*LATENT] supernodes_in
#define WS_ZEND    (WS_SNIN + NC*LATENT)     // zero range end
#define WS_MEANS   WS_ZEND                   // [NC*EMB]
#define WS_SNODES  (WS_MEANS + NC*EMB)       // [NC*LATENT] supernodes
#define WS_EMBN    (WS_SNODES + NC*LATENT)   // [N*EMB]    normalized emb
#define WS_RCPL1   (WS_EMBN + 100000*EMB)    // [N]
#define WS_BLIKE   (WS_RCPL1 + 100000)       // [8N]  likelihood -> weight
#define WS_SSRC    (WS_BLIKE + 800000)       // [20480] int
#define WS_SDST    (WS_SSRC + 20480)         // [20480] int
#define WS_BDST    (WS_SDST + 20480)         // [8N] int

// ---------------------------------------------------------------------------
__global__ void k_zero(float* p, int n) {
  int i = blockIdx.x * blockDim.x + threadIdx.x;
  if (i < n) p[i] = 0.f;
}

// emb normalize + scatter into cluster sums + nodes L1 reciprocal
__global__ void k_node_prep(const float* __restrict__ emb,
                            const float* __restrict__ nodes,
                            const int* __restrict__ clusters,
                            float* __restrict__ emb_n,
                            float* __restrict__ rcp_l1,
                            float* __restrict__ csums,
                            float* __restrict__ counts, int N) {
  int i = blockIdx.x * blockDim.x + threadIdx.x;
  if (i >= N) return;
  float e[EMB]; float ss = 0.f;
#pragma unroll
  for (int j = 0; j < EMB; ++j) { e[j] = emb[i*EMB + j]; ss += e[j]*e[j]; }
  float inv = 1.f / (sqrtf(ss) + EPSF);
  int c = clusters[i];
#pragma unroll
  for (int j = 0; j < EMB; ++j) {
    float v = e[j] * inv;
    emb_n[i*EMB + j] = v;
    atomicAdd(&csums[c*EMB + j], v);
  }
  atomicAdd(&counts[c], 1.f);
  float s1 = 0.f;
  for (int j = 0; j < LATENT; j += 4) {
    float4 v = *(const float4*)&nodes[i*LATENT + j];
    s1 += fabsf(v.x) + fabsf(v.y) + fabsf(v.z) + fabsf(v.w);
  }
  rcp_l1[i] = 1.f / (s1 + EPSF);
}

// cluster means = normalize(sums / max(count,1)); also seed supernodes[:, :16]
__global__ void k_means(const float* __restrict__ csums,
                        const float* __restrict__ counts,
                        float* __restrict__ means,
                        float* __restrict__ snodes) {
  int r = blockIdx.x * blockDim.x + threadIdx.x;
  if (r >= NC) return;
  float inv = 1.f / fmaxf(counts[r], 1.f);
  float m[EMB]; float ss = 0.f;
#pragma unroll
  for (int j = 0; j < EMB; ++j) { m[j] = csums[r*EMB + j] * inv; ss += m[j]*m[j]; }
  float nrm = 1.f / (sqrtf(ss) + EPSF);
#pragma unroll
  for (int j = 0; j < EMB; ++j) {
    float v = m[j] * nrm;
    means[r*EMB + j] = v;
    snodes[r*LATENT + j] = v;   // concat([means, mlp(...)]) first 16 cols
  }
}

// super graph: top-10 kNN among 1024 means, symmetric edge list
__global__ __launch_bounds__(256) void k_super_knn(const float* __restrict__ means,
                                                   int* __restrict__ s_src,
                                                   int* __restrict__ s_dst) {
  __shared__ float sm[NC*EMB];              // 64 KB
  int tid = threadIdx.x;
  for (int t = tid; t < NC*EMB; t += 256) sm[t] = means[t];
  __syncthreads();
  int r = blockIdx.x * 256 + tid;           // grid = 4 blocks, r in [0,1024)
  float e[EMB];
#pragma unroll
  for (int j = 0; j < EMB; ++j) e[j] = sm[r*EMB + j];
  float vals[K_SUPER]; int idxs[K_SUPER];
#pragma unroll
  for (int j = 0; j < K_SUPER; ++j) { vals[j] = -3.4e38f; idxs[j] = 0; }
  for (int c = 0; c < NC; ++c) {
    float s = 0.f;
#pragma unroll
    for (int j = 0; j < EMB; ++j) s += e[j] * sm[c*EMB + j];
    float cv = s; int ci = c;                 // register insertion network
#pragma unroll
    for (int j = 0; j < K_SUPER; ++j) {
      bool sw = cv > vals[j];
      float tv = sw ? vals[j] : cv; int ti = sw ? idxs[j] : ci;
      vals[j] = sw ? cv : vals[j]; idxs[j] = sw ? ci : idxs[j];
      cv = tv; ci = ti;
    }
  }
#pragma unroll
  for (int j = 0; j < K_SUPER; ++j) {
    int e0 = r*K_SUPER + j;
    s_src[e0] = r;              s_dst[e0] = idxs[j];
    s_src[NC*K_SUPER + e0] = idxs[j]; s_dst[NC*K_SUPER + e0] = r;  // symmetric
  }
}

// bipartite: top-8 of each node vs means + BN batch stats (wave-reduced)
__global__ __launch_bounds__(256) void k_bip_knn(const float* __restrict__ emb_n,
                                                 const float* __restrict__ means,
                                                 int* __restrict__ b_dst,
                                                 float* __restrict__ b_like,
                                                 float* __restrict__ bns, int N) {
  __shared__ float sm[NC*EMB];              // 64 KB
  int tid = threadIdx.x;
  for (int t = tid; t < NC*EMB; t += 256) sm[t] = means[t];
  __syncthreads();
  int i = blockIdx.x * 256 + tid;
  float lsum = 0.f, lsq = 0.f;
  if (i < N) {
    float e[EMB];
#pragma unroll
    for (int j = 0; j < EMB; ++j) e[j] = emb_n[i*EMB + j];
    float vals[K_BIP]; int idxs[K_BIP];
#pragma unroll
    for (int j = 0; j < K_BIP; ++j) { vals[j] = -3.4e38f; idxs[j] = 0; }
    for (int c = 0; c < NC; ++c) {
      float s = 0.f;
#pragma unroll
      for (int j = 0; j < EMB; ++j) s += e[j] * sm[c*EMB + j];
      float cv = s; int ci = c;
#pragma unroll
      for (int j = 0; j < K_BIP; ++j) {
        bool sw = cv > vals[j];
        float tv = sw ? vals[j] : cv; int ti = sw ? idxs[j] : ci;
        vals[j] = sw ? cv : vals[j]; idxs[j] = sw ? ci : idxs[j];
        cv = tv; ci = ti;
      }
    }
#pragma unroll
    for (int j = 0; j < K_BIP; ++j) {
      b_dst[i*K_BIP + j]  = idxs[j];
      b_like[i*K_BIP + j] = vals[j];
      lsum += vals[j]; lsq += vals[j]*vals[j];
    }
  }
  for (int o = 16; o > 0; o >>= 1) {        // wave32 reduction
    lsum += __shfl_down(lsum, o, 32);
    lsq  += __shfl_down(lsq,  o, 32);
  }
  if ((tid & 31) == 0) { atomicAdd(&bns[0], lsum); atomicAdd(&bns[1], lsq); }
}

// BatchNorm -> exp -> denom accumulate (w overwrites b_like)
__global__ void k_bip_w(const int* __restrict__ b_dst, float* __restrict__ b_like,
                        const float* __restrict__ bns,
                        const float* __restrict__ g, const float* __restrict__ b,
                        float* __restrict__ b_den, int E) {
  int e = blockIdx.x * blockDim.x + threadIdx.x;
  if (e >= E) return;
  float invE = 1.f / (float)E;
  float mean = bns[0] * invE;
  float var  = bns[1] * invE - mean*mean;
  float rstd = rsqrtf(var + 1e-5f);
  float logit = (b_like[e] - mean) * rstd * g[0] + b[0];
  float w = expf(logit);
  b_like[e] = w;
  atomicAdd(&b_den[b_dst[e]], w);
}

// supernodes_in[dst] += nodes_l1[src] * w   (one thread per edge-quarter)
__global__ void k_scatter(const int* __restrict__ b_dst,
                          const float* __restrict__ b_w,
                          const float* __restrict__ b_den,
                          const float* __restrict__ nodes,
                          const float* __restrict__ rcp_l1,
                          float* __restrict__ sn_in, int E) {
  int t = blockIdx.x * blockDim.x + threadIdx.x;
  if (t >= E * 4) return;
  int e = t >> 2, q = t & 3;
  int node = e >> 3;               // b_src = repeat(arange(N), 8)
  int d = b_dst[e];
  float w  = b_w[e] / (b_den[d] + EPSF);
  float sc = w * rcp_l1[node];
  int f0 = q * 32;
  for (int f = f0; f < f0 + 32; f += 4) {
    float4 v = *(const float4*)&nodes[node*LATENT + f];
    atomicAdd(&sn_in[d*LATENT + f + 0], v.x * sc);
    atomicAdd(&sn_in[d*LATENT + f + 1], v.y * sc);
    atomicAdd(&sn_in[d*LATENT + f + 2], v.z * sc);
    atomicAdd(&sn_in[d*LATENT + f + 3], v.w * sc);
  }
}

// ---------------------------------------------------------------------------
// Fused 2-layer MLP: out = relu(ln(relu(ln(A@W0+b0)) @ W1 + b1))
// 16-row block per workgroup, 128 threads (4 waves), V_WMMA_F32_16X16X4_F32.
// Optional gather mode: A row r = concat(gbase[gsrc[r]], gbase[gdst[r]]).
// ---------------------------------------------------------------------------
__global__ __launch_bounds__(128) void k_mlp2(
    const float* __restrict__ Ain,
    const int* __restrict__ gsrc, const int* __restrict__ gdst,
    const float* __restrict__ gbase,
    const float* __restrict__ W0, const float* __restrict__ b0,
    const float* __restrict__ g0, const float* __restrict__ be0,
    const float* __restrict__ W1, const float* __restrict__ b1,
    const float* __restrict__ g1, const float* __restrict__ be1,
    float* __restrict__ out, int out_stride, int out_off,
    int K1, int k1shift, int N2)
{
  __shared__ float sA[16*256];      // input tile (K1 <= 256)
  __shared__ float sH[16*HIDDEN];   // hidden activations
  __shared__ float sC[16*128];      // layer-2 output tile (N2 <= 128)
  __shared__ float sRed[16*16];
  __shared__ float sStat[16*2];
  const int tid  = threadIdx.x;
  const int wave = tid >> 5, lane = tid & 31;
  const int m16  = lane & 15;
  const int kh   = (lane >> 4) << 1;       // K split: lanes 0-15 -> {k0,k0+1}, 16-31 -> {k0+2,k0+3}
  const int rbase = (lane >> 4) * 8;       // C/D: M split at lane 16 (ISA 7.12.2)
  const int row0 = blockIdx.x * 16;

  // ---- load A tile (optionally gathered pair-concat rows) ----
  for (int t = tid; t < 16*K1; t += 128) {
    int r = t >> k1shift, c = t & (K1 - 1);
    float v;
    if (gsrc) {
      int row = row0 + r;
      int idx = (c < LATENT) ? gsrc[row] : gdst[row];
      v = gbase[idx*LATENT + (c & (LATENT-1))];
    } else {
      v = Ain[(row0 + r)*K1 + c];
    }
    sA[t] = v;
  }
  __syncthreads();

  // ---- GEMM1: sH = sA @ W0 + b0  (N = HIDDEN, 32 tiles over 4 waves) ----
  for (int nt = wave; nt < HIDDEN/16; nt += 4) {
    int n = nt*16 + m16;
    v8f acc = {0.f,0.f,0.f,0.f,0.f,0.f,0.f,0.f};
    for (int k0 = 0; k0 < K1; k0 += 4) {
      v2f a = { sA[m16*K1 + k0 + kh], sA[m16*K1 + k0 + kh + 1] };
      v2f bb = { W0[(k0 + kh)*HIDDEN + n], W0[(k0 + kh + 1)*HIDDEN + n] };
      acc = __builtin_amdgcn_wmma_f32_16x16x4_f32(false, a, false, bb,
                                                  (short)0, acc, false, false);
    }
    float bias = b0[n];
#pragma unroll
    for (int j = 0; j < 8; ++j) sH[(rbase + j)*HIDDEN + n] = acc[j] + bias;
  }
  __syncthreads();

  // ---- LayerNorm + ReLU over HIDDEN, in place (8 lanes per row) ----
  {
    int r = tid >> 3, l8 = tid & 7;
    float s = 0.f, sq = 0.f;
    for (int c = l8; c < HIDDEN; c += 8) { float x = sH[r*HIDDEN + c]; s += x; sq += x*x; }
    sRed[r*16 + l8] = s; sRed[r*16 + 8 + l8] = sq;
    __syncthreads();
    if (l8 == 0) {
      float S = 0.f, Q = 0.f;
#pragma unroll
      for (int j = 0; j < 8; ++j) { S += sRed[r*16 + j]; Q += sRed[r*16 + 8 + j]; }
      float mean = S * (1.f/HIDDEN);
      float var  = Q * (1.f/HIDDEN) - mean*mean;
      sStat[r*2] = mean; sStat[r*2 + 1] = rsqrtf(var + 1e-5f);
    }
    __syncthreads();
    float mean = sStat[r*2], rstd = sStat[r*2 + 1];
    for (int c = l8; c < HIDDEN; c += 8) {
      float x = (sH[r*HIDDEN + c] - mean) * rstd * g0[c] + be0[c];
      sH[r*HIDDEN + c] = fmaxf(x, 0.f);
    }
  }
  __syncthreads();

  // ---- GEMM2: sC = sH @ W1 + b1 ----
  for (int nt = wave; nt < (N2 >> 4); nt += 4) {
    int n = nt*16 + m16;
    v8f acc = {0.f,0.f,0.f,0.f,0.f,0.f,0.f,0.f};
    for (int k0 = 0; k0 < HIDDEN; k0 += 4) {
      v2f a = { sH[m16*HIDDEN + k0 + kh], sH[m16*HIDDEN + k0 + kh + 1] };
      v2f bb = { W1[(k0 + kh)*N2 + n], W1[(k0 + kh + 1)*N2 + n] };
      acc = __builtin_amdgcn_wmma_f32_16x16x4_f32(false, a, false, bb,
                                                  (short)0, acc, false, false);
    }
    float bias = b1[n];
#pragma unroll
    for (int j = 0; j < 8; ++j) sC[(rbase + j)*N2 + n] = acc[j] + bias;
  }
  __syncthreads();

  // ---- LayerNorm + ReLU over N2 -> global ----
  {
    int r = tid >> 3, l8 = tid & 7;
    float s = 0.f, sq = 0.f;
    for (int c = l8; c < N2; c += 8) { float x = sC[r*N2 + c]; s += x; sq += x*x; }
    sRed[r*16 + l8] = s; sRed[r*16 + 8 + l8] = sq;
    __syncthreads();
    if (l8 == 0) {
      float S = 0.f, Q = 0.f;
#pragma unroll
      for (int j = 0; j < 8; ++j) { S += sRed[r*16 + j]; Q += sRed[r*16 + 8 + j]; }
      float mean = S / (float)N2;
      float var  = Q / (float)N2 - mean*mean;
      sStat[r*2] = mean; sStat[r*2 + 1] = rsqrtf(var + 1e-5f);
    }
    __syncthreads();
    float mean = sStat[r*2], rstd = sStat[r*2 + 1];
    for (int c = l8; c < N2; c += 8) {
      float x = (sC[r*N2 + c] - mean) * rstd * g1[c] + be1[c];
      out[(row0 + r)*out_stride + out_off + c] = fmaxf(x, 0.f);
    }
  }
}

// ---------------------------------------------------------------------------
extern "C" void kernel_launch(void* const* d_in, const int* in_sizes, int n_in,
                              void* d_out, int out_size, void* d_ws, size_t ws_size,
                              hipStream_t stream) {
  const float* embeddings = (const float*)d_in[1];
  const float* nodes      = (const float*)d_in[2];
  const float* sn_w0 = (const float*)d_in[4];
  const float* sn_b0 = (const float*)d_in[5];
  const float* sn_g0 = (const float*)d_in[6];
  const float* sn_be0= (const float*)d_in[7];
  const float* sn_w1 = (const float*)d_in[8];
  const float* sn_b1 = (const float*)d_in[9];
  const float* sn_g1 = (const float*)d_in[10];
  const float* sn_be1= (const float*)d_in[11];
  const float* se_w0 = (const float*)d_in[12];
  const float* se_b0 = (const float*)d_in[13];
  const float* se_g0 = (const float*)d_in[14];
  const float* se_be0= (const float*)d_in[15];
  const float* se_w1 = (const float*)d_in[16];
  const float* se_b1 = (const float*)d_in[17];
  const float* se_g1 = (const float*)d_in[18];
  const float* se_be1= (const float*)d_in[19];
  const float* bg_gamma = (const float*)d_in[22];
  const float* bg_beta  = (const float*)d_in[23];
  const int*   clusters = (const int*)d_in[26];
  const int N = in_sizes[26];                    // 100000 nodes
  const int EB = N * K_BIP;                      // 800000 bipartite edges
  const int MSE = 2 * NC * K_SUPER;              // 20480 superedges

  float* ws = (float*)d_ws;
  float* csums  = ws + WS_CSUMS;
  float* counts = ws + WS_CNT;
  float* bns    = ws + WS_BNS;
  float* b_den  = ws + WS_BDEN;
  float* sn_in  = ws + WS_SNIN;
  float* means  = ws + WS_MEANS;
  float* snodes = ws + WS_SNODES;
  float* emb_n  = ws + WS_EMBN;
  float* rcp_l1 = ws + WS_RCPL1;
  float* b_like = ws + WS_BLIKE;
  int*   s_src  = (int*)(ws + WS_SSRC);
  int*   s_dst  = (int*)(ws + WS_SDST);
  int*   b_dst  = (int*)(ws + WS_BDST);

  // 1) zero accumulators (required every call: harness poisons ws once)
  k_zero<<<(WS_ZEND + 255)/256, 256, 0, stream>>>(ws, WS_ZEND);
  // 2) emb normalize + cluster scatter + nodes L1 rcp
  k_node_prep<<<(N + 255)/256, 256, 0, stream>>>(embeddings, nodes, clusters,
                                                 emb_n, rcp_l1, csums, counts, N);
  // 3) cluster means (+ seed supernodes[:, :16])
  k_means<<<NC/256, 256, 0, stream>>>(csums, counts, means, snodes);
  // 4) super graph top-10 (symmetric edge list)
  k_super_knn<<<NC/256, 256, 0, stream>>>(means, s_src, s_dst);
  // 5) bipartite top-8 + BN stats
  k_bip_knn<<<(N + 255)/256, 256, 0, stream>>>(emb_n, means, b_dst, b_like, bns, N);
  // 6) BN -> exp -> per-cluster denom
  k_bip_w<<<(EB + 255)/256, 256, 0, stream>>>(b_dst, b_like, bns, bg_gamma, bg_beta,
                                              b_den, EB);
  // 7) weighted scatter into supernodes_in
  k_scatter<<<(EB*4 + 255)/256, 256, 0, stream>>>(b_dst, b_like, b_den, nodes,
                                                  rcp_l1, sn_in, EB);
  // 8) supernode MLP: 128 -> 512 -> 112, write cols 16..127 of supernodes
  k_mlp2<<<NC/16, 128, 0, stream>>>(sn_in, (const int*)nullptr, (const int*)nullptr,
                                    (const float*)nullptr,
                                    sn_w0, sn_b0, sn_g0, sn_be0,
                                    sn_w1, sn_b1, sn_g1, sn_be1,
                                    snodes, LATENT, EMB,
                                    LATENT, 7, LATENT - EMB);
  // 9) superedge MLP (gathered pair rows): 256 -> 512 -> 128 -> d_out
  k_mlp2<<<MSE/16, 128, 0, stream>>>((const float*)nullptr, s_src, s_dst, snodes,
                                     se_w0, se_b0, se_g0, se_be0,
                                     se_w1, se_b1, se_g1, se_be1,
                                     (float*)d_out, LATENT, 0,
                                     2*LATENT, 8, LATENT);
  (void)n_in; (void)out_size; (void)ws_size; (void)in_sizes;
}